// GCN_36189394436507
// MI455X (gfx1250) — compile-verified
//
#include <hip/hip_runtime.h>

#define BN_EPS 1e-5f

typedef float v2f __attribute__((ext_vector_type(2)));
typedef float v8f __attribute__((ext_vector_type(8)));

// ---------------------------------------------------------------------------
// Degree accumulation: deg[src[e]] += 1  (fp32 atomics, L2-resident)
// ---------------------------------------------------------------------------
__global__ void deg_kernel(float* __restrict__ deg, const int* __restrict__ src, int E) {
  int e = blockIdx.x * blockDim.x + threadIdx.x;
  if (e < E) atomicAdd(&deg[src[e]], 1.0f);
}

// dis[i] = deg>0 ? rsqrt(deg) : 0   (in place; max(deg,1)==deg when deg>0)
__global__ void dis_kernel(float* __restrict__ deg, int n) {
  int i = blockIdx.x * blockDim.x + threadIdx.x;
  if (i < n) {
    float d = deg[i];
    deg[i] = (d > 0.0f) ? rsqrtf(d) : 0.0f;
  }
}

// w[e] = -dis[src]*dis[dst]
__global__ void w_kernel(float* __restrict__ w, const float* __restrict__ dis,
                         const int* __restrict__ src, const int* __restrict__ dst, int E) {
  int e = blockIdx.x * blockDim.x + threadIdx.x;
  if (e < E) w[e] = -dis[src[e]] * dis[dst[e]];
}

// Build padded input: X[n, 0:16] = [x(3) | pos(3) | normals(3) | zeros(7)]
__global__ void concat_kernel(float* __restrict__ X, const float* __restrict__ x,
                              const float* __restrict__ pos, const float* __restrict__ nrm, int n) {
  int i = blockIdx.x * blockDim.x + threadIdx.x;
  if (i >= n) return;
  float* r = X + i * 16;
  r[0] = x[3 * i];   r[1] = x[3 * i + 1];   r[2] = x[3 * i + 2];
  r[3] = pos[3 * i]; r[4] = pos[3 * i + 1]; r[5] = pos[3 * i + 2];
  r[6] = nrm[3 * i]; r[7] = nrm[3 * i + 1]; r[8] = nrm[3 * i + 2];
#pragma unroll
  for (int j = 9; j < 16; ++j) r[j] = 0.0f;
}

// Zero-pad W1 [3,9,64] -> [3,16,64] so the WMMA K-loop needs no guards.
__global__ void padw_kernel(float* __restrict__ Wp, const float* __restrict__ W,
                            int finLog, int finPad, int fout) {
  int i = blockIdx.x * blockDim.x + threadIdx.x;
  int total = 3 * finPad * fout;
  if (i >= total) return;
  int f = i % fout;
  int k = (i / fout) % finPad;
  int t = i / (fout * finPad);
  Wp[i] = (k < finLog) ? W[(t * finLog + k) * fout + f] : 0.0f;
}

// Y = -X  (Tx2 accumulator init: Tx2 = 2*lap(Tx1) - h)
__global__ void neg_init(float* __restrict__ Y, const float* __restrict__ X, int n) {
  int i = blockIdx.x * blockDim.x + threadIdx.x;
  if (i < n) Y[i] = -X[i];
}

// ---------------------------------------------------------------------------
// Edge-parallel SpMM scatter: Y[dst[e], :] += scale * w[e] * X[src[e], :]
// groups is a power of two -> shift/mask addressing, no integer division.
// 16B feature groups per thread; atomics land in the 192MB L2.
// ---------------------------------------------------------------------------
__global__ void spmm_scatter(float* __restrict__ Y, const float* __restrict__ X,
                             const int* __restrict__ src, const int* __restrict__ dst,
                             const float* __restrict__ w, int total, int gshift,
                             int Fpad, float scale) {
  int t = blockIdx.x * blockDim.x + threadIdx.x;
  if (t >= total) return;
  int e  = t >> gshift;
  int fg = t & ((1 << gshift) - 1);
  int s = src[e];
  int d = dst[e];
  float we = scale * w[e];
  const float4 xv = *(const float4*)(X + s * Fpad + (fg << 2));
  float* yp = Y + d * Fpad + (fg << 2);
  atomicAdd(yp + 0, we * xv.x);
  atomicAdd(yp + 1, we * xv.y);
  atomicAdd(yp + 2, we * xv.z);
  atomicAdd(yp + 3, we * xv.w);
}

// ---------------------------------------------------------------------------
// Fused ChebConv dense part, fp32 WMMA (V_WMMA_F32_16X16X4_F32):
//   out = PReLU(X0@W0 + T1@W1 + T2@W2 + b), plus BN sum/sumsq accumulation.
// Weight (B) fragments for all 3 taps preloaded into registers once per wave,
// reused across `rowTiles` 16-row tiles (one layer's taps = 96 VGPRs, fits).
// ---------------------------------------------------------------------------
template <int FINPAD, int FOUT>
__global__ void cheb_gemm(const float* __restrict__ X0, const float* __restrict__ T1,
                          const float* __restrict__ T2, const float* __restrict__ W,
                          const float* __restrict__ bias, const float* __restrict__ alpha,
                          float* __restrict__ out, float* __restrict__ stats, int rowTiles) {
  constexpr int KS = FINPAD / 4;
  const int lane = threadIdx.x;
  const int colBase = blockIdx.y * 16;
  const int m    = lane & 15;
  const int koff = (lane >> 4) << 1;  // 0 for lanes 0-15, 2 for lanes 16-31
  const int rowOff = (lane >> 4) * 8;

  v2f b0[KS], b1[KS], b2[KS];
  const float* w0 = W;
  const float* w1 = W + FINPAD * FOUT;
  const float* w2 = W + 2 * FINPAD * FOUT;
#pragma unroll
  for (int ks = 0; ks < KS; ++ks) {
    int kk = 4 * ks + koff;
    int wi = kk * FOUT + colBase + m;
    b0[ks].x = w0[wi]; b0[ks].y = w0[wi + FOUT];
    b1[ks].x = w1[wi]; b1[ks].y = w1[wi + FOUT];
    b2[ks].x = w2[wi]; b2[ks].y = w2[wi + FOUT];
  }
  const float al = alpha[0];
  const float bc = bias[colBase + m];
  const v8f vzero = {0.f, 0.f, 0.f, 0.f, 0.f, 0.f, 0.f, 0.f};

  for (int rt = 0; rt < rowTiles; ++rt) {
    const int rowBase = (blockIdx.x * rowTiles + rt) * 16;
    const int rb = (rowBase + m) * FINPAD;
    const float* a0p = X0 + rb;
    const float* a1p = T1 + rb;
    const float* a2p = T2 + rb;
    v8f c = vzero;
#pragma unroll
    for (int ks = 0; ks < KS; ++ks) {
      int kk = 4 * ks + koff;
      float2 f0 = *(const float2*)(a0p + kk);
      float2 f1 = *(const float2*)(a1p + kk);
      float2 f2 = *(const float2*)(a2p + kk);
      v2f a;
      a.x = f0.x; a.y = f0.y;
      c = __builtin_amdgcn_wmma_f32_16x16x4_f32(false, a, false, b0[ks], (short)0, c, false, false);
      a.x = f1.x; a.y = f1.y;
      c = __builtin_amdgcn_wmma_f32_16x16x4_f32(false, a, false, b1[ks], (short)0, c, false, false);
      a.x = f2.x; a.y = f2.y;
      c = __builtin_amdgcn_wmma_f32_16x16x4_f32(false, a, false, b2[ks], (short)0, c, false, false);
    }
    float s = 0.0f, s2 = 0.0f;
#pragma unroll
    for (int r = 0; r < 8; ++r) {
      float v = c[r] + bc;
      v = (v > 0.0f) ? v : al * v;  // PReLU
      out[(rowBase + rowOff + r) * FOUT + colBase + m] = v;
      s += v;
      s2 += v * v;
    }
    s  += __shfl_xor(s, 16, 32);
    s2 += __shfl_xor(s2, 16, 32);
    if (lane < 16) {
      atomicAdd(&stats[colBase + m], s);
      atomicAdd(&stats[FOUT + colBase + m], s2);
    }
  }
}

// BatchNorm apply: dst = (src - mu)/sqrt(var+eps)*g + be  (biased var; Fout pow2)
__global__ void bn_apply(const float* __restrict__ src, const float* __restrict__ stats,
                         const float* __restrict__ g, const float* __restrict__ be,
                         float* __restrict__ dst, int total, int fmask, int Fout, float invN) {
  int i = blockIdx.x * blockDim.x + threadIdx.x;
  if (i >= total) return;
  int f = i & fmask;
  float mu  = stats[f] * invN;
  float var = fmaxf(stats[Fout + f] * invN - mu * mu, 0.0f);
  dst[i] = (src[i] - mu) * rsqrtf(var + BN_EPS) * g[f] + be[f];
}

// ---------------------------------------------------------------------------
// Fully fused MLP head: out = (relu(relu(X@W1+b1)@W2+b2))@W3+b3, F=32.
// All three 32x32 weight matrices staged in LDS (12KB) -- keeps register
// pressure low (no spills); B fragments come from ds loads. Inter-layer
// activations staged via 16x32 LDS tiles. A-fragments prefetched as a clause.
// ---------------------------------------------------------------------------
__global__ void mlp_fused(const float* __restrict__ X,
                          const float* __restrict__ W1, const float* __restrict__ B1v,
                          const float* __restrict__ W2, const float* __restrict__ B2v,
                          const float* __restrict__ W3, const float* __restrict__ B3v,
                          float* __restrict__ out, int rowTiles) {
  constexpr int F = 32;
  constexpr int KS = 8;
  __shared__ float wlds[3 * F * F];   // 12KB: three 32x32 weight matrices
  __shared__ float tileA[16 * F];
  __shared__ float tileB[16 * F];
  const int lane = threadIdx.x;
  const int m    = lane & 15;
  const int koff = (lane >> 4) << 1;
  const int rowOff = (lane >> 4) * 8;

  // cooperative weight staging: 3*1024 floats, float4 per lane per step
  {
    const float* Ws[3] = {W1, W2, W3};
#pragma unroll
    for (int l = 0; l < 3; ++l)
      for (int i = lane * 4; i < F * F; i += 32 * 4)
        *(float4*)&wlds[l * F * F + i] = *(const float4*)&Ws[l][i];
  }
  float bv[3][2];
  bv[0][0] = B1v[m]; bv[0][1] = B1v[16 + m];
  bv[1][0] = B2v[m]; bv[1][1] = B2v[16 + m];
  bv[2][0] = B3v[m]; bv[2][1] = B3v[16 + m];
  __syncthreads();
  const v8f vzero = {0.f, 0.f, 0.f, 0.f, 0.f, 0.f, 0.f, 0.f};

  for (int rt = 0; rt < rowTiles; ++rt) {
    const int rowBase = (blockIdx.x * rowTiles + rt) * 16;

    // ---- layer 1: A from global (prefetch 8 fragments as a clause) ----
    const float* ap = X + (rowBase + m) * F;
    float2 af[KS];
#pragma unroll
    for (int ks = 0; ks < KS; ++ks) af[ks] = *(const float2*)(ap + 4 * ks + koff);
    v8f c0 = vzero, c1 = vzero;
#pragma unroll
    for (int ks = 0; ks < KS; ++ks) {
      int kk = 4 * ks + koff;
      v2f a, b;
      a.x = af[ks].x; a.y = af[ks].y;
      b.x = wlds[kk * F + m];          b.y = wlds[(kk + 1) * F + m];
      c0 = __builtin_amdgcn_wmma_f32_16x16x4_f32(false, a, false, b, (short)0, c0, false, false);
      b.x = wlds[kk * F + 16 + m];     b.y = wlds[(kk + 1) * F + 16 + m];
      c1 = __builtin_amdgcn_wmma_f32_16x16x4_f32(false, a, false, b, (short)0, c1, false, false);
    }
    __syncthreads();  // previous iteration's tileA reads complete
#pragma unroll
    for (int r = 0; r < 8; ++r) {
      tileA[(rowOff + r) * F + m]      = fmaxf(c0[r] + bv[0][0], 0.0f);
      tileA[(rowOff + r) * F + 16 + m] = fmaxf(c1[r] + bv[0][1], 0.0f);
    }
    __syncthreads();

    // ---- layer 2: A from LDS tileA ----
#pragma unroll
    for (int ks = 0; ks < KS; ++ks) af[ks] = *(const float2*)(&tileA[m * F + 4 * ks + koff]);
    c0 = vzero; c1 = vzero;
#pragma unroll
    for (int ks = 0; ks < KS; ++ks) {
      int kk = 4 * ks + koff;
      v2f a, b;
      a.x = af[ks].x; a.y = af[ks].y;
      b.x = wlds[F * F + kk * F + m];      b.y = wlds[F * F + (kk + 1) * F + m];
      c0 = __builtin_amdgcn_wmma_f32_16x16x4_f32(false, a, false, b, (short)0, c0, false, false);
      b.x = wlds[F * F + kk * F + 16 + m]; b.y = wlds[F * F + (kk + 1) * F + 16 + m];
      c1 = __builtin_amdgcn_wmma_f32_16x16x4_f32(false, a, false, b, (short)0, c1, false, false);
    }
    __syncthreads();
#pragma unroll
    for (int r = 0; r < 8; ++r) {
      tileB[(rowOff + r) * F + m]      = fmaxf(c0[r] + bv[1][0], 0.0f);
      tileB[(rowOff + r) * F + 16 + m] = fmaxf(c1[r] + bv[1][1], 0.0f);
    }
    __syncthreads();

    // ---- layer 3: A from LDS tileB, write logits to global ----
#pragma unroll
    for (int ks = 0; ks < KS; ++ks) af[ks] = *(const float2*)(&tileB[m * F + 4 * ks + koff]);
    c0 = vzero; c1 = vzero;
#pragma unroll
    for (int ks = 0; ks < KS; ++ks) {
      int kk = 4 * ks + koff;
      v2f a, b;
      a.x = af[ks].x; a.y = af[ks].y;
      b.x = wlds[2 * F * F + kk * F + m];      b.y = wlds[2 * F * F + (kk + 1) * F + m];
      c0 = __builtin_amdgcn_wmma_f32_16x16x4_f32(false, a, false, b, (short)0, c0, false, false);
      b.x = wlds[2 * F * F + kk * F + 16 + m]; b.y = wlds[2 * F * F + (kk + 1) * F + 16 + m];
      c1 = __builtin_amdgcn_wmma_f32_16x16x4_f32(false, a, false, b, (short)0, c1, false, false);
    }
#pragma unroll
    for (int r = 0; r < 8; ++r) {
      out[(rowBase + rowOff + r) * F + m]      = c0[r] + bv[2][0];
      out[(rowBase + rowOff + r) * F + 16 + m] = c1[r] + bv[2][1];
    }
    __syncthreads();  // tileB safe to overwrite next iteration
  }
}

// ---------------------------------------------------------------------------
extern "C" void kernel_launch(void* const* d_in, const int* in_sizes, int n_in,
                              void* d_out, int out_size, void* d_ws, size_t ws_size,
                              hipStream_t stream) {
  const float* x   = (const float*)d_in[0];
  const float* pos = (const float*)d_in[1];
  const float* nrm = (const float*)d_in[2];
  const int*   ei  = (const int*)d_in[3];
  const int N = in_sizes[0] / 3;   // 100000 (divisible by 16; N/16 divisible by 5)
  const int E = in_sizes[3] / 2;   // 3200000
  const int* src = ei;
  const int* dst = ei + E;

  // workspace carve-up (floats)
  float* ws   = (float*)d_ws;
  float* dis  = ws;              ws += N;               // deg -> dis in place
  float* wE   = ws;              ws += E;               // edge weights
  float* B0   = ws;              ws += (size_t)N * 64;  // layer input (padded)
  float* B1   = ws;              ws += (size_t)N * 64;  // Tx1
  float* B2   = ws;              ws += (size_t)N * 64;  // Tx2
  float* B3   = ws;              ws += (size_t)N * 64;  // pre-BN output
  float* stats = ws;             ws += 128;             // [sum | sumsq]
  float* Wpad  = ws;             ws += 3 * 16 * 64;     // padded W1

  const int TB = 256;
  const int RPW = 5;                       // row tiles per wave (6250 % 5 == 0)
  const int GX = (N / 16) / RPW;           // 1250
  auto cdiv = [](int a, int b) { return (a + b - 1) / b; };

  hipMemsetAsync(dis, 0, (size_t)N * 4, stream);
  deg_kernel<<<cdiv(E, TB), TB, 0, stream>>>(dis, src, E);
  dis_kernel<<<cdiv(N, TB), TB, 0, stream>>>(dis, N);
  w_kernel<<<cdiv(E, TB), TB, 0, stream>>>(wE, dis, src, dst, E);
  concat_kernel<<<cdiv(N, TB), TB, 0, stream>>>(B0, x, pos, nrm, N);
  padw_kernel<<<cdiv(3 * 16 * 64, TB), TB, 0, stream>>>(Wpad, (const float*)d_in[4], 9, 16, 64);

  for (int L = 0; L < 4; ++L) {
    const int FinPad = (L == 0) ? 16 : 64;
    const int Fout   = (L == 3) ? 32 : 64;
    const float* Wl  = (L == 0) ? Wpad : (const float*)d_in[4 + 5 * L];
    const float* bl  = (const float*)d_in[5 + 5 * L];
    const float* al  = (const float*)d_in[6 + 5 * L];
    const float* gl  = (const float*)d_in[7 + 5 * L];
    const float* bel = (const float*)d_in[8 + 5 * L];

    const int total = N * FinPad;
    hipMemsetAsync(B1, 0, (size_t)total * 4, stream);            // Tx1 = 0
    neg_init<<<cdiv(total, TB), TB, 0, stream>>>(B2, B0, total); // Tx2 = -h
    const int gshift = (FinPad == 16) ? 2 : 4;                   // groups = FinPad/4
    const int sTot = E << gshift;
    spmm_scatter<<<cdiv(sTot, TB), TB, 0, stream>>>(B1, B0, src, dst, wE, sTot, gshift, FinPad, 1.0f);
    spmm_scatter<<<cdiv(sTot, TB), TB, 0, stream>>>(B2, B1, src, dst, wE, sTot, gshift, FinPad, 2.0f);
    hipMemsetAsync(stats, 0, (size_t)2 * Fout * 4, stream);
    if (L == 0) {
      cheb_gemm<16, 64><<<dim3(GX, 4), 32, 0, stream>>>(B0, B1, B2, Wl, bl, al, B3, stats, RPW);
    } else if (L == 3) {
      cheb_gemm<64, 32><<<dim3(GX, 2), 32, 0, stream>>>(B0, B1, B2, Wl, bl, al, B3, stats, RPW);
    } else {
      cheb_gemm<64, 64><<<dim3(GX, 4), 32, 0, stream>>>(B0, B1, B2, Wl, bl, al, B3, stats, RPW);
    }
    bn_apply<<<cdiv(N * Fout, TB), TB, 0, stream>>>(B3, stats, gl, bel, B0, N * Fout,
                                                    Fout - 1, Fout, 1.0f / (float)N);
  }

  // Fused MLP head: 32 -> relu -> 32 -> relu -> 32 (logits to d_out)
  mlp_fused<<<GX, 32, 0, stream>>>(B0,
                                   (const float*)d_in[24], (const float*)d_in[25],
                                   (const float*)d_in[26], (const float*)d_in[27],
                                   (const float*)d_in[28], (const float*)d_in[29],
                                   (float*)d_out, RPW);
}